// GraphEmbeddingModel_3779571221004
// MI455X (gfx1250) — compile-verified
//
#include <hip/hip_runtime.h>
#include <hip/hip_bf16.h>

typedef __attribute__((ext_vector_type(2))) float v2f;
typedef __attribute__((ext_vector_type(8))) float v8f;

#define D 128
#define SW_STRIDE 144   // W tile LDS row stride (floats): 2*144 % 64-banks = 32 -> half-waves hit disjoint bank ranges
#define SA_STRIDE 132   // A tile LDS row stride (floats): bank = (4*row + k) % 64 -> conflict-free b64 reads

// ---------- degree / normalization ----------
__global__ __launch_bounds__(256) void k_deg_init(float* __restrict__ deg, int n) {
    int i = blockIdx.x * 256 + threadIdx.x;
    if (i < n) deg[i] = 1.0f;  // self-loop contributes 1
}

__global__ __launch_bounds__(256) void k_deg_count(const int* __restrict__ dst,
                                                   float* __restrict__ deg, int e) {
    int i = blockIdx.x * 256 + threadIdx.x;
    if (i < e) atomicAdd(&deg[dst[i]], 1.0f);
}

__global__ __launch_bounds__(256) void k_deg_rsqrt(float* __restrict__ deg, int n) {
    int i = blockIdx.x * 256 + threadIdx.x;
    if (i < n) deg[i] = rsqrtf(deg[i]);
}

// ---------- fp32 WMMA GEMM: H[n,128] = op(A)[n,128] @ W[128,128] ----------
// Block = 256 threads = 8 waves. LDS-staged: A tile (16x128) loaded once and
// shared by all 8 waves; full W (128x128) loaded once per block. Inner loop is
// ds_load (A: b64, W: 2x b32) feeding v_wmma_f32_16x16x4_f32, 32 steps.
template <bool RELU_IN, bool USE_IDX>
__global__ __launch_bounds__(256) void k_gemm_wmma(const float* __restrict__ A,
                                                   const int* __restrict__ idx,
                                                   const float* __restrict__ W,
                                                   float* __restrict__ H, int n) {
    extern __shared__ float smem[];
    float* sW = smem;                    // [128][SW_STRIDE]
    float* sA = smem + 128 * SW_STRIDE;  // [16][SA_STRIDE]

    const int t    = threadIdx.x;
    const int lane = t & 31;
    const int wv   = t >> 5;
    const int m0   = blockIdx.x << 4;
    const int mr   = lane & 15;
    const int kh   = (lane >> 4) << 1;        // 0 or 2 (K half within a 4-step)
    const int ncol = (wv << 4) + mr;          // output column 0..127

    // --- stage W: 128x128 floats, 4096 float4 loads across 256 threads ---
#pragma unroll
    for (int i = 0; i < 16; ++i) {
        int j = t + i * 256;                  // float4 index
        int r = j >> 5, c = (j & 31) << 2;    // row, col
        float4 w4 = *(const float4*)(W + r * D + c);
        *(float4*)(sW + r * SW_STRIDE + c) = w4;
    }
    // --- stage A tile: 16x128 floats, 1024 float2 loads (idx gather + optional relu) ---
#pragma unroll
    for (int i = 0; i < 4; ++i) {
        int f = t + i * 256;                  // float2 index
        int r = f >> 6, c = (f & 63) << 1;    // tile row 0..15, col
        int gr = m0 + r;
        if (gr >= n) gr = n - 1;              // clamp (EXEC stays full)
        if (USE_IDX) gr = idx[gr];
        float2 a2 = *(const float2*)(A + (size_t)gr * D + c);
        if (RELU_IN) { a2.x = fmaxf(a2.x, 0.0f); a2.y = fmaxf(a2.y, 0.0f); }
        *(float2*)(sA + r * SA_STRIDE + c) = a2;
    }
    __syncthreads();

    v8f c = {};
#pragma unroll 8
    for (int kt = 0; kt < D; kt += 4) {
        const int k = kt + kh;
        v2f a = *(const v2f*)(sA + mr * SA_STRIDE + k);
        v2f b;
        b.x = sW[k * SW_STRIDE + ncol];
        b.y = sW[(k + 1) * SW_STRIDE + ncol];
        // (neg_a, A, neg_b, B, c_mod, C, reuse_a, reuse_b)
        c = __builtin_amdgcn_wmma_f32_16x16x4_f32(false, a, false, b,
                                                  (short)0, c, false, false);
    }

    const int rbase = m0 + ((lane >> 4) << 3);
    float* Hp = H + (size_t)rbase * D + ncol;
    if (m0 + 16 <= n) {
        // fast path: whole stripe in range, 8 unguarded stores off one base
#pragma unroll
        for (int r = 0; r < 8; ++r) Hp[r * D] = c[r];
    } else {
#pragma unroll
        for (int r = 0; r < 8; ++r)
            if (rbase + r < n) Hp[r * D] = c[r];
    }
}

// ---------- acc[i,:] = h[i,:] * dinv[i]^2 + bias  (self-loop term, full overwrite) ----------
__global__ __launch_bounds__(256) void k_init_acc(const float* __restrict__ h,
                                                  const float* __restrict__ dinv,
                                                  const float* __restrict__ bias,
                                                  float* __restrict__ acc, int n) {
    long long gid = (long long)blockIdx.x * 256 + threadIdx.x;
    if (gid >= (long long)n * 32) return;
    int i = (int)(gid >> 5), q = (int)(gid & 31);
    float di = dinv[i];
    float w  = di * di;
    float4 v = *(const float4*)(h + (size_t)i * D + q * 4);
    float4 b = *(const float4*)(bias + q * 4);
    float4 o;
    o.x = v.x * w + b.x; o.y = v.y * w + b.y;
    o.z = v.z * w + b.z; o.w = v.w * w + b.w;
    *(float4*)(acc + (size_t)i * D + q * 4) = o;
}

// ---------- edge scatter: acc[dst,:] += h[src,:] * dinv[src]*dinv[dst] ----------
// 32 threads per edge, float4 per thread; gathers are 512B-contiguous per edge,
// atomics resolve in L2 (h + acc working set fits in MI455X's 192MB L2).
__global__ __launch_bounds__(256) void k_edge_scatter(const int* __restrict__ src,
                                                      const int* __restrict__ dst,
                                                      const float* __restrict__ h,
                                                      const float* __restrict__ dinv,
                                                      float* acc, int e) {
    long long gid = (long long)blockIdx.x * 256 + threadIdx.x;
    if (gid >= (long long)e * 32) return;
    int ei = (int)(gid >> 5), q = (int)(gid & 31);
    int s = src[ei], d = dst[ei];
    float w = dinv[s] * dinv[d];
    float4 v = *(const float4*)(h + (size_t)s * D + q * 4);
    float* p = acc + (size_t)d * D + q * 4;
    atomicAdd(p + 0, v.x * w);
    atomicAdd(p + 1, v.y * w);
    atomicAdd(p + 2, v.z * w);
    atomicAdd(p + 3, v.w * w);
}

extern "C" void kernel_launch(void* const* d_in, const int* in_sizes, int n_in,
                              void* d_out, int out_size, void* d_ws, size_t ws_size,
                              hipStream_t stream) {
    const int*   x_idx = (const int*)d_in[0];
    const int*   eidx  = (const int*)d_in[1];
    const float* emb   = (const float*)d_in[2];
    const float* W1    = (const float*)d_in[3];
    const float* b1    = (const float*)d_in[4];
    const float* W2    = (const float*)d_in[5];
    const float* b2    = (const float*)d_in[6];
    float* out = (float*)d_out;

    const int N = in_sizes[0];
    const int E = in_sizes[1] / 2;
    const int* src = eidx;       // edge_index[0]
    const int* dst = eidx + E;   // edge_index[1]

    // workspace layout: dinv[N] | h[N*128] | xacc[N*128]
    char*  ws   = (char*)d_ws;
    size_t off  = ((size_t)N * 4 + 255) & ~(size_t)255;
    float* dinv = (float*)ws;
    float* h    = (float*)(ws + off);
    float* xacc = (float*)(ws + off + (size_t)N * D * 4);

    const int gN     = (N + 255) / 256;
    const int gE     = (E + 255) / 256;
    const int gTiles = (N + 15) / 16;
    const int gN32   = (int)(((long long)N * 32 + 255) / 256);
    const int gE32   = (int)(((long long)E * 32 + 255) / 256);
    const size_t lds = (size_t)(128 * SW_STRIDE + 16 * SA_STRIDE) * sizeof(float); // ~80 KB of the 320 KB WGP LDS

    // symmetric deg^-1/2 normalization (shared by both layers)
    k_deg_init <<<gN, 256, 0, stream>>>(dinv, N);
    k_deg_count<<<gE, 256, 0, stream>>>(dst, dinv, E);
    k_deg_rsqrt<<<gN, 256, 0, stream>>>(dinv, N);

    // layer 1: h1 = emb[x_idx] @ W1 ; xacc = scatter(h1) + self-loop + b1
    k_gemm_wmma<false, true><<<gTiles, 256, lds, stream>>>(emb, x_idx, W1, h, N);
    k_init_acc<<<gN32, 256, 0, stream>>>(h, dinv, b1, xacc, N);
    k_edge_scatter<<<gE32, 256, 0, stream>>>(src, dst, h, dinv, xacc, E);

    // layer 2: h2 = relu(xacc) @ W2 ; out = scatter(h2) + self-loop + b2
    k_gemm_wmma<true, false><<<gTiles, 256, lds, stream>>>(xacc, nullptr, W2, h, N);
    k_init_acc<<<gN32, 256, 0, stream>>>(h, dinv, b2, out, N);
    k_edge_scatter<<<gE32, 256, 0, stream>>>(src, dst, h, dinv, out, E);
}